// KVLOhmGNNLayer_12678743458331
// MI455X (gfx1250) — compile-verified
//
#include <hip/hip_runtime.h>

// -----------------------------------------------------------------------------
// KVL Ohm GNN layer for MI455X (gfx1250, wave32).
//
//  Phase 1: zero the per-node net-current accumulator in d_ws (800 KB).
//  Phase 2: edge kernel — gather V2, complex multiply, write I_edge/V_edge,
//           scatter-add net current with hardware f32 atomics (L2-resident).
//  Phase 3: node GEMM — (N x 130) @ (130 x 128) + bias, ReLU, using
//           V_WMMA_F32_16X16X4_F32. One workgroup (8 waves) = 16x128 out tile.
//           K-loop: guard-free 32 WMMA steps (K=0..127, unrolled x4 so loads
//           pipeline ahead of the matrix ops) + 1 branchless epilogue step
//           covering K=128..131 (cols 130/131 zero-padded).
// -----------------------------------------------------------------------------

typedef __attribute__((ext_vector_type(2))) float v2f;
typedef __attribute__((ext_vector_type(8))) float v8f;

#define D_FEAT   128
#define OUT_DIM  128
#define K_FULL   130
#define K_PAD    132   // K rounded up to multiple of 4 for 16x16x4 WMMA

// ---------------------------------------------------------------- zero ws ----
__global__ void kvl_zero_kernel(float* __restrict__ p, int n) {
    int i = blockIdx.x * blockDim.x + threadIdx.x;
    if (i < n) p[i] = 0.0f;
}

// -------------------------------------------------------------- edge pass ----
__global__ void kvl_edge_kernel(const float* __restrict__ V_node,
                                const int*   __restrict__ senders,
                                const int*   __restrict__ receivers,
                                const float* __restrict__ ef,     // (E,2) G,B
                                float* __restrict__ I_edge,       // (E,2)
                                float* __restrict__ V_edge,       // (E,2)
                                float* __restrict__ net,          // (N,2) accum
                                int n_edges) {
    int e = blockIdx.x * blockDim.x + threadIdx.x;
    if (e >= n_edges) return;

    int s = senders[e];
    int r = receivers[e];

    // V2 gathers: first two columns of the 128-wide feature rows.
    // The 800 KB V2 footprint is L2-resident on MI455X (192 MB L2).
    float2 vr = *(const float2*)(V_node + (size_t)r * D_FEAT);
    float2 vs = *(const float2*)(V_node + (size_t)s * D_FEAT);
    float vre = vr.x - vs.x;
    float vim = vr.y - vs.y;

    float2 gb = *(const float2*)(ef + (size_t)e * 2);
    float ire = gb.x * vre - gb.y * vim;
    float iim = gb.x * vim + gb.y * vre;

    *(float2*)(V_edge + (size_t)e * 2) = make_float2(vre, vim);
    *(float2*)(I_edge + (size_t)e * 2) = make_float2(ire, iim);

    // net_current = I_in - I_out: +I at receiver, -I at sender.
    // unsafeAtomicAdd lowers to global_atomic_add_f32 (no CAS loop).
    unsafeAtomicAdd(net + (size_t)r * 2 + 0,  ire);
    unsafeAtomicAdd(net + (size_t)r * 2 + 1,  iim);
    unsafeAtomicAdd(net + (size_t)s * 2 + 0, -ire);
    unsafeAtomicAdd(net + (size_t)s * 2 + 1, -iim);
}

// ---------------------------------------------------------- node GEMM+ReLU ---
// Block: 256 threads (8 wave32). Block b computes rows [16b, 16b+16).
// Wave w computes output columns [16w, 16w+16) via V_WMMA_F32_16X16X4_F32.
__global__ void __launch_bounds__(256)
kvl_node_gemm_kernel(const float* __restrict__ V_node,   // (N,128)
                     const float* __restrict__ net,      // (N,2)
                     const float* __restrict__ W,        // (130,128)
                     const float* __restrict__ bias,     // (128)
                     float* __restrict__ out,            // (N,128)
                     int n_nodes) {
    __shared__ float As[16][K_PAD];   // 16 x 132 f32 = 8448 B, conflict-free

    const int m0   = blockIdx.x * 16;
    const int tid  = threadIdx.x;
    const int lane = tid & 31;        // wave32
    const int wave = tid >> 5;        // 0..7
    const int half = lane >> 4;       // 0 | 1 (lane group)
    const int l16  = lane & 15;
    const int n0   = wave * 16;

    const bool full_tile = (m0 + 16 <= n_nodes);

    // ---- Stage A tile: node features (cols 0..127) as float4 ----
    for (int i = tid; i < 16 * (D_FEAT / 4); i += 256) {
        int row = i >> 5;             // /32
        int c4  = i & 31;
        int g   = m0 + row;
        float4 v = make_float4(0.f, 0.f, 0.f, 0.f);
        if (g < n_nodes)
            v = *(const float4*)(V_node + (size_t)g * D_FEAT + c4 * 4);
        *(float4*)&As[row][c4 * 4] = v;
    }
    // ---- cols 128..129 = net current, 130..131 = zero pad ----
    if (tid < 64) {
        int row = tid >> 2;           // 0..15
        int col = tid & 3;            // 0..3
        int g   = m0 + row;
        float v = 0.0f;
        if (col < 2 && g < n_nodes) v = net[(size_t)g * 2 + col];
        As[row][D_FEAT + col] = v;
    }
    __syncthreads();

    // Per-lane K offset inside each 4-wide step:
    //   A/B frags: VGPR0 -> K = kbase + 2*half, VGPR1 -> K = kbase + 2*half+1.
    const int koff = 2 * half;
    const float* __restrict__ Wcol = W + n0 + l16;   // this lane's W column

    v8f acc = {};

    // ---- Main K loop: K = 0..127, no bounds checks, pipelined x4 ----
    #pragma unroll 4
    for (int kk = 0; kk < 32; ++kk) {
        const int k0 = kk * 4 + koff;
        v2f a, b;
        a.x = As[l16][k0 + 0];
        a.y = As[l16][k0 + 1];
        b.x = Wcol[(size_t)(k0 + 0) * OUT_DIM];
        b.y = Wcol[(size_t)(k0 + 1) * OUT_DIM];
        acc = __builtin_amdgcn_wmma_f32_16x16x4_f32(
            false, a, false, b, (short)0, acc, false, false);
    }

    // ---- Epilogue step: K = 128..131 (rows 130/131 of W do not exist) ----
    {
        const int k0 = 128 + koff;                 // 128/129 or 130/131
        v2f a, b;
        a.x = As[l16][k0 + 0];                     // LDS already zero-padded
        a.y = As[l16][k0 + 1];
        const int kx0 = (k0 + 0 < K_FULL) ? (k0 + 0) : 0;   // safe address
        const int kx1 = (k0 + 1 < K_FULL) ? (k0 + 1) : 0;
        float w0 = Wcol[(size_t)kx0 * OUT_DIM];
        float w1 = Wcol[(size_t)kx1 * OUT_DIM];
        b.x = (k0 + 0 < K_FULL) ? w0 : 0.0f;       // v_cndmask, no EXEC branch
        b.y = (k0 + 1 < K_FULL) ? w1 : 0.0f;
        acc = __builtin_amdgcn_wmma_f32_16x16x4_f32(
            false, a, false, b, (short)0, acc, false, false);
    }

    // ---- bias + ReLU + store; D layout: M = j + 8*half, N = l16 ----
    const float bb = bias[n0 + l16];
    float* __restrict__ outp =
        out + (size_t)(m0 + 8 * half) * OUT_DIM + n0 + l16;

    if (full_tile) {                               // uniform (scalar) branch
        #pragma unroll
        for (int j = 0; j < 8; ++j) {
            float v = acc[j] + bb;
            outp[(size_t)j * OUT_DIM] = v > 0.0f ? v : 0.0f;
        }
    } else {
        #pragma unroll
        for (int j = 0; j < 8; ++j) {
            int row = m0 + j + 8 * half;
            if (row < n_nodes) {
                float v = acc[j] + bb;
                outp[(size_t)j * OUT_DIM] = v > 0.0f ? v : 0.0f;
            }
        }
    }
}

// ------------------------------------------------------------------ launch ---
extern "C" void kernel_launch(void* const* d_in, const int* in_sizes, int n_in,
                              void* d_out, int out_size, void* d_ws, size_t ws_size,
                              hipStream_t stream) {
    const float* V_node    = (const float*)d_in[0];   // (N,128)
    const int*   senders   = (const int*)  d_in[1];   // (E)
    const int*   receivers = (const int*)  d_in[2];   // (E)
    const float* edge_feat = (const float*)d_in[3];   // (E,2)
    const float* W         = (const float*)d_in[4];   // (130,128)
    const float* bias      = (const float*)d_in[5];   // (128)

    const int n_nodes = in_sizes[0] / D_FEAT;
    const int n_edges = in_sizes[1];

    // Tuple output layout, flat-concatenated in return order:
    //   V_node_out (N,128) | I_edge (E,2) | V_edge (E,2)
    float* out_nodes = (float*)d_out;
    float* out_Iedge = out_nodes + (size_t)n_nodes * OUT_DIM;
    float* out_Vedge = out_Iedge + (size_t)n_edges * 2;

    float* net = (float*)d_ws;                        // (N,2) accumulator

    // Phase 1: zero net-current accumulator (must happen every call).
    {
        int n = n_nodes * 2;
        kvl_zero_kernel<<<(n + 255) / 256, 256, 0, stream>>>(net, n);
    }
    // Phase 2: edge compute + scatter-add.
    {
        int blocks = (n_edges + 255) / 256;
        kvl_edge_kernel<<<blocks, 256, 0, stream>>>(
            V_node, senders, receivers, edge_feat,
            out_Iedge, out_Vedge, net, n_edges);
    }
    // Phase 3: WMMA GEMM + bias + ReLU.
    {
        int blocks = (n_nodes + 15) / 16;
        kvl_node_gemm_kernel<<<blocks, 256, 0, stream>>>(
            V_node, net, W, bias, out_nodes, n_nodes);
    }
}